// LlamaAttention_27702539059452
// MI455X (gfx1250) — compile-verified
//
#include <hip/hip_runtime.h>
#include <math.h>

// ---------------- problem constants (match reference) ----------------
constexpr int B_   = 64;
constexpr int H_   = 32;
constexpr int KVH_ = 8;
constexpr int D_   = 128;
constexpr int HID_ = 4096;
constexpr int QKVD = (H_ + 2 * KVH_) * D_;   // 6144
constexpr int BS_  = 16;                      // keys per page == WMMA tile M
constexpr int MB_  = 128;                     // pages per sequence
constexpr int S_   = BS_ * MB_;               // 2048
constexpr float SCALE_ = 0.08838834764831845f; // D^-0.5
constexpr float THETA_ = 10000.0f;

typedef unsigned int u32;
typedef __attribute__((ext_vector_type(2))) float v2f;
typedef __attribute__((ext_vector_type(8))) float v8f;
typedef __attribute__((ext_vector_type(4))) unsigned int v4u;
typedef __attribute__((ext_vector_type(4))) int v4i;
typedef __attribute__((ext_vector_type(8))) int v8i;

// D = A(16x4,f32) x B(4x16,f32) + C(16x16,f32)   -- native CDNA5 fp32 WMMA
__device__ __forceinline__ v8f wmma_f32(v2f a, v2f b, v8f c) {
  return __builtin_amdgcn_wmma_f32_16x16x4_f32(
      /*neg_a=*/false, a, /*neg_b=*/false, b,
      /*c_mod=*/(short)0, c, /*reuse_a=*/false, /*reuse_b=*/false);
}

// ---------------- TDM staging parameters for the GEMM ----------------
constexpr int KC      = 128;       // K elements staged per TDM transfer
constexpr int WROWS   = 64;        // weight rows (N) per block tile
constexpr int WSTRIDE = KC + 2;    // padded LDS row stride in DWORDs
                                   // (TDM pads 2 DW per 512B row: conflict-free
                                   //  transposed reads, 8B alignment kept)

// Issue one TDM 2D tile load: WROWS x KC fp32 tile, row stride `ld` elements,
// into LDS at byte offset lds_off with per-row padding.
// D# per CDNA5 ISA (group0 128b, group1 256b; 2D so
// groups 2/3 zero). Descriptor is workgroup-uniform.
// clang-23 toolchain: 6-arg builtin (v4u, v8i, v4i, v4i, v8i, i32 cpol).
__device__ __forceinline__ void tdm_load_wtile(const float* g, u32 lds_off, int ld) {
  unsigned long long ga = (unsigned long long)(size_t)g;
  v4u g0;
  g0[0] = 1u;                                  // count=1, user desc, no gather
  g0[1] = lds_off;                             // lds_addr (bytes)
  g0[2] = (u32)ga;                             // global_addr[31:0]
  g0[3] = (u32)(ga >> 32) | (2u << 30);        // global_addr[56:32] | type=2
  v8i g1;
  g1[0] = (2 << 16)                            // data_size = 4 bytes
        | (1 << 20)                            // pad_enable
        | (6 << 22)                            // pad_interval: every 64x8B = 512B
        | (1 << 25);                           // pad_amount: 2 DWORDs
  g1[1] = (int)((u32)KC << 16);                // tensor_dim0[15:0] (=KC)
  g1[2] = (int)((u32)WROWS << 16);             // dim0 hi=0 | tensor_dim1[15:0]
  g1[3] = (int)((u32)KC << 16);                // dim1 hi=0 | tile_dim0 = KC
  g1[4] = WROWS;                               // tile_dim1 = WROWS, tile_dim2 = 0
  g1[5] = ld;                                  // tensor_dim0_stride[31:0]
  g1[6] = 0;                                   // stride hi | dim1_stride lo (2D)
  g1[7] = 0;
  v4i g2 = (v4i)0;
  v4i g3 = (v4i)0;
  v8i g4 = (v8i)0;                             // unused trailing group (zeroed)
  __builtin_amdgcn_tensor_load_to_lds(g0, g1, g2, g3, g4, 0);
}

// =====================================================================
// C[M,N] = A[M,K] * W[N,K]^T   (row-major; weights stored (out,in)).
// Block = 4 waves: tile M=16, N=64; wave wv owns columns wv*16..wv*16+15.
// W tiles (64 x 128 fp32, 32KB) stream HBM->LDS via the Tensor Data
// Mover, double-buffered, TENSORcnt-synchronized. A rows read directly
// (L2-resident). WMMA operands:
//   A: lane(l16,h): {A[m0+l16][k+2h], A[m0+l16][k+2h+1]}
//   B: lane(l16,h): {Wlds[n-local][k+2h], +1}   (padded stride 130 DW)
//   C: vgpr r, lane: C[m0 + r + 8*half][n0 + l16]
// =====================================================================
__global__ __launch_bounds__(128) void gemm_nt_wmma(
    const float* __restrict__ A, const float* __restrict__ W,
    float* __restrict__ C, int N, int K) {
  __shared__ float w_lds[2][WROWS * WSTRIDE];  // 2 x 33.3 KB

  const int lane = threadIdx.x & 31;
  const int wv   = threadIdx.x >> 5;
  const int half = lane >> 4;
  const int l16  = lane & 15;
  const int n0 = blockIdx.x * 64 + wv * 16;
  const int m0 = blockIdx.y * 16;
  const bool issuer = (wv == 0);

  const float* wbase = W + (size_t)(blockIdx.x * 64) * K;   // 64-row N tile
  const u32 lds0 = (u32)(size_t)&w_lds[0][0];
  const u32 lds1 = (u32)(size_t)&w_lds[1][0];

  if (issuer) tdm_load_wtile(wbase, lds0, K);               // prologue: chunk 0

  const float* ap = A + (size_t)(m0 + l16) * K + 2 * half;
  const float* wl = &w_lds[0][(wv * 16 + l16) * WSTRIDE + 2 * half];

  v8f c;
#pragma unroll
  for (int r = 0; r < 8; ++r) c[r] = 0.0f;

  for (int kc = 0; kc < K; kc += KC) {
    const int cur = (kc / KC) & 1;
    if (issuer) {
      if (kc + KC < K) {                       // prefetch next chunk, keep 1 inflight
        tdm_load_wtile(wbase + (kc + KC), cur ? lds0 : lds1, K);
        __builtin_amdgcn_s_wait_tensorcnt((short)1);
      } else {
        __builtin_amdgcn_s_wait_tensorcnt((short)0);
      }
    }
    __syncthreads();                           // LDS buffer cur is resident for all waves

    const float* apk = ap + kc;
    const float* wp  = wl + cur * (WROWS * WSTRIDE);
#pragma unroll
    for (int kk = 0; kk < KC; kk += 4) {
      v2f a = *(const v2f*)(apk + kk);
      v2f b = *(const v2f*)(wp + kk);
      c = wmma_f32(a, b, c);
    }
    __syncthreads();                           // done reading before refill
  }

#pragma unroll
  for (int r = 0; r < 8; ++r)
    C[(size_t)(m0 + r + 8 * half) * N + n0 + l16] = c[r];
}

// =====================================================================
// In-place RoPE on q (scaled by D^-1/2) and k rows of the qkv scratch.
// grid (B, H+KVH), block 64 (one thread per rotation pair).
// =====================================================================
__global__ __launch_bounds__(64) void rope_kernel(
    float* __restrict__ qkv, const int* __restrict__ positions) {
  const int b  = blockIdx.x;
  const int hh = blockIdx.y;            // 0..H-1 => q head, H..H+KVH-1 => k head
  const int i  = threadIdx.x;           // pair index 0..63
  float* row = qkv + (size_t)b * QKVD + (size_t)hh * D_;

  const float pos = (float)positions[b];
  const float inv = __powf(THETA_, -(float)(2 * i) / (float)D_);
  float sn, cs;
  __sincosf(pos * inv, &sn, &cs);

  float x1 = row[i], x2 = row[i + D_ / 2];
  float o1 = x1 * cs - x2 * sn;
  float o2 = x2 * cs + x1 * sn;
  if (hh < H_) { o1 *= SCALE_; o2 *= SCALE_; }   // fold softmax scale into q
  row[i] = o1;
  row[i + D_ / 2] = o2;
}

// =====================================================================
// Paged decode attention, one workgroup per (batch, kv-head).
// 4 waves x 512 keys, online softmax, WMMA for QK^T and PV.
// The just-written KV slot (positions[b]) is substituted from qkv
// scratch so the input caches are never mutated.
// =====================================================================
__global__ __launch_bounds__(128) void paged_attn_wmma(
    const float* __restrict__ qkv, const float* __restrict__ kc,
    const float* __restrict__ vc, const int* __restrict__ positions,
    const int* __restrict__ btab, const int* __restrict__ ctxlen,
    float* __restrict__ attn_out) {
  constexpr int NW = 4;
  constexpr int KPW = S_ / NW;                 // 512 keys per wave

  __shared__ float q_lds[16 * D_];             // 4 real q rows, 12 zero rows (8 KB)
  __shared__ float p_lds[NW][16 * 16];         // probs, [key][g] per wave (4 KB)
  __shared__ float r_m[NW][4], r_l[NW][4];
  __shared__ float r_acc[NW][4][D_];           // per-wave partial outputs (8 KB)

  const int b    = blockIdx.x;
  const int kv   = blockIdx.y;
  const int tid  = threadIdx.x;
  const int lane = tid & 31;
  const int wv   = tid >> 5;
  const int half = lane >> 4;
  const int l16  = lane & 15;

  const int pos = positions[b];
  const int ctx = ctxlen[b];

  // q heads kv*4 .. kv*4+3 are contiguous (512 floats); zero-pad to 16 rows
  const float* qsrc = qkv + (size_t)b * QKVD + (size_t)(kv * 4) * D_;
  for (int i = tid; i < 16 * D_; i += 128)
    q_lds[i] = (i < 4 * D_) ? qsrc[i] : 0.0f;
  __syncthreads();

  const float* knew = qkv + (size_t)b * QKVD + (size_t)(H_ + kv) * D_;
  const float* vnew = qkv + (size_t)b * QKVD + (size_t)(H_ + KVH_ + kv) * D_;

  float m[4], lsum[4];
#pragma unroll
  for (int g = 0; g < 4; ++g) { m[g] = -1e30f; lsum[g] = 0.0f; }

  v8f cpv[8];                                  // out[g][d], 8 d-tiles of 16
#pragma unroll
  for (int t = 0; t < 8; ++t)
#pragma unroll
    for (int r = 0; r < 8; ++r) cpv[t][r] = 0.0f;

  const int kbase = wv * KPW;
  int nkeys = ctx - kbase;
  nkeys = nkeys < 0 ? 0 : (nkeys > KPW ? KPW : nkeys);
  const int nchunks = (nkeys + 15) >> 4;

  for (int ch = 0; ch < nchunks; ++ch) {
    const int c0 = kbase + ch * 16;            // page-aligned: one page per chunk
    const int page = btab[b * MB_ + (c0 >> 4)];
    const int s_lane = c0 + l16;               // this lane's key row for QK
    const float* krow = (s_lane == pos)
        ? knew
        : kc + (((size_t)page * BS_ + l16) * KVH_ + kv) * D_;

    // ---- scores^T (key x head) = K_tile x Q^T, K-dim = D in steps of 4 ----
    v8f cqk;
#pragma unroll
    for (int r = 0; r < 8; ++r) cqk[r] = 0.0f;
#pragma unroll
    for (int ds = 0; ds < 32; ++ds) {
      const int d = ds * 4 + 2 * half;
      v2f a  = *(const v2f*)(krow + d);
      v2f bq = *(const v2f*)(&q_lds[l16 * D_ + d]);
      cqk = wmma_f32(a, bq, cqk);
    }

    // ---- mask + online softmax (head g lives in lanes {g, g+16}) ----
    float cmax = -1e30f;
#pragma unroll
    for (int r = 0; r < 8; ++r) {
      const int key = c0 + r + 8 * half;
      if (key >= ctx) cqk[r] = -1e30f;
      cmax = fmaxf(cmax, cqk[r]);
    }
    cmax = fmaxf(cmax, __shfl_xor(cmax, 16, 32));

    float mnew[4], fac[4];
#pragma unroll
    for (int g = 0; g < 4; ++g) {
      const float mg = __shfl(cmax, g, 32);
      mnew[g] = fmaxf(m[g], mg);
      fac[g]  = __expf(m[g] - mnew[g]);
    }

    float psum = 0.0f;
#pragma unroll
    for (int r = 0; r < 8; ++r) {
      const int key = c0 + r + 8 * half;
      const float p = (key < ctx) ? __expf(cqk[r] - mnew[l16 & 3]) : 0.0f;
      p_lds[wv][(r + 8 * half) * 16 + l16] = p;   // [key][g]
      psum += p;
    }
    psum += __shfl_xor(psum, 16, 32);
#pragma unroll
    for (int g = 0; g < 4; ++g) {
      const float sg = __shfl(psum, g, 32);
      lsum[g] = lsum[g] * fac[g] + sg;
      m[g] = mnew[g];
    }

    // rescale accumulators (rows r>=4 / upper half are dead padding)
#pragma unroll
    for (int t = 0; t < 8; ++t)
#pragma unroll
      for (int r = 0; r < 8; ++r) cpv[t][r] *= fac[r & 3];

    // ---- out += P(g x 16keys) x V(16keys x d), 4 key-steps x 8 d-tiles ----
#pragma unroll
    for (int js = 0; js < 4; ++js) {
      const int jj = js * 4 + 2 * half;        // this lane's two V rows
      const int s0 = c0 + jj, s1 = s0 + 1;
      const float* vr0 = (s0 == pos)
          ? vnew : vc + (((size_t)page * BS_ + jj) * KVH_ + kv) * D_;
      const float* vr1 = (s1 == pos)
          ? vnew : vc + (((size_t)page * BS_ + jj + 1) * KVH_ + kv) * D_;
      v2f a;
      a.x = p_lds[wv][jj * 16 + l16];
      a.y = p_lds[wv][(jj + 1) * 16 + l16];
#pragma unroll
      for (int t = 0; t < 8; ++t) {
        v2f bv;
        bv.x = vr0[t * 16 + l16];
        bv.y = vr1[t * 16 + l16];
        cpv[t] = wmma_f32(a, bv, cpv[t]);
      }
    }
  }

  // ---- stash per-wave partials, then merge across the 4 waves ----
  if (lane == 0)
#pragma unroll
    for (int g = 0; g < 4; ++g) { r_m[wv][g] = m[g]; r_l[wv][g] = lsum[g]; }
  if (half == 0) {
#pragma unroll
    for (int t = 0; t < 8; ++t)
#pragma unroll
      for (int r = 0; r < 4; ++r) r_acc[wv][r][t * 16 + l16] = cpv[t][r];
  }
  __syncthreads();

  for (int i = tid; i < 4 * D_; i += 128) {
    const int g = i / D_, d = i % D_;
    float M = -1e30f;
    for (int w = 0; w < NW; ++w) M = fmaxf(M, r_m[w][g]);
    float den = 0.0f, num = 0.0f;
    for (int w = 0; w < NW; ++w) {
      const float e = __expf(r_m[w][g] - M);
      den += e * r_l[w][g];
      num += e * r_acc[w][g][d];
    }
    attn_out[((size_t)b * H_ + kv * 4 + g) * D_ + d] = num / den;
  }
}

// =====================================================================
extern "C" void kernel_launch(void* const* d_in, const int* in_sizes, int n_in,
                              void* d_out, int out_size, void* d_ws, size_t ws_size,
                              hipStream_t stream) {
  const float* hs   = (const float*)d_in[0];
  const float* Wqkv = (const float*)d_in[1];
  const float* Wo   = (const float*)d_in[2];
  const float* kcache = (const float*)d_in[3];
  const float* vcache = (const float*)d_in[4];
  const int* positions = (const int*)d_in[5];
  const int* btab      = (const int*)d_in[6];
  const int* ctx       = (const int*)d_in[7];
  float* out = (float*)d_out;

  float* ws_qkv  = (float*)d_ws;                       // B x 6144
  float* ws_attn = ws_qkv + (size_t)B_ * QKVD;         // B x 4096

  // 1) qkv = hidden @ Wqkv^T   (TDM-staged 100 MB weight stream)
  gemm_nt_wmma<<<dim3(QKVD / 64, B_ / 16), 128, 0, stream>>>(
      hs, Wqkv, ws_qkv, QKVD, HID_);

  // 2) RoPE(q,k), q pre-scaled by D^-1/2
  rope_kernel<<<dim3(B_, H_ + KVH_), 64, 0, stream>>>(ws_qkv, positions);

  // 3) paged attention (memory-bound on ~1 GB of KV cache)
  paged_attn_wmma<<<dim3(B_, KVH_), 128, 0, stream>>>(
      ws_qkv, kcache, vcache, positions, btab, ctx, ws_attn);

  // 4) out = attn @ Wo^T   (TDM-staged 64 MB weight stream)
  gemm_nt_wmma<<<dim3(HID_ / 64, B_ / 16), 128, 0, stream>>>(
      ws_attn, Wo, out, HID_, HID_);
}